// FreLoss_67877663146258
// MI455X (gfx1250) — compile-verified
//
#include <hip/hip_runtime.h>
#include <math.h>

typedef __attribute__((ext_vector_type(2))) float v2f;
typedef __attribute__((ext_vector_type(8))) float v8f;
typedef __attribute__((ext_vector_type(4))) int   v4i;

#define NLAT  512
#define NLON  1024
#define MGRID (NLAT * NLON)   // 524288
#define NPTS  2048
#define LMAX  50
#define MMAX  50
#define NBB   4               // 2 branches x 2 batches
#define PI_F  3.14159265358979323846f
#define PI_D  3.14159265358979323846

#if __has_builtin(__builtin_amdgcn_global_load_async_to_lds_b128)
#define HAVE_ASYNC_LDS 1
#else
#define HAVE_ASYNC_LDS 0
#endif

typedef __attribute__((address_space(1))) v4i gas_v4i;
typedef __attribute__((address_space(3))) v4i las_v4i;

// ---------------------------------------------------------------------------
// Stage 1: spherical conversion + packing of the WMMA "A" rows.
// A row for cloud point p: [-2*theta, -2*phi, 1, theta^2+phi^2]
// ---------------------------------------------------------------------------
__global__ void prep_kernel(const float* __restrict__ pred,
                            const float* __restrict__ target,
                            float4* __restrict__ pts4,
                            float* __restrict__ rr) {
  int idx = blockIdx.x * 256 + threadIdx.x;
  if (idx >= NBB * NPTS) return;
  int bb = idx / NPTS;            // bb = branch*2 + batch
  int n  = idx % NPTS;
  const float* src = ((bb < 2) ? pred : target) + ((size_t)(bb & 1) * NPTS + n) * 3;
  float x = src[0], y = src[1], z = src[2];
  float r   = sqrtf(x * x + y * y + z * z);
  float nyz = sqrtf(y * y + z * z);
  float th  = acosf(x / r);
  float base = acosf(y / nyz);
  float ph  = ((z < 0.0f) ? (2.0f * PI_F - base) : base) - PI_F;
  pts4[idx] = make_float4(-2.0f * th, -2.0f * ph, 1.0f, th * th + ph * ph);
  rr[idx]   = r;
}

// ---------------------------------------------------------------------------
// Clenshaw-Curtis weights for n=512 (N=511, all b_k = 2).
// ---------------------------------------------------------------------------
__global__ void wq_kernel(float* __restrict__ wq) {
  int j = blockIdx.x * 256 + threadIdx.x;
  if (j >= NLAT) return;
  double t = PI_D * (double)j / 511.0;
  double S = 0.0;
  for (int k = 1; k <= 255; ++k)
    S += cos(2.0 * (double)k * t) * (2.0 / (4.0 * (double)k * (double)k - 1.0));
  double c = (j == 0 || j == NLAT - 1) ? 1.0 : 2.0;
  wq[j] = (float)((c / 511.0) * (1.0 - S));
}

// ---------------------------------------------------------------------------
// Orthonormalized associated Legendre P[l][m][k], recurrence in f64.
// One thread per (m, k); writes all l (zeros for l < m).
// ---------------------------------------------------------------------------
__global__ void plm_kernel(float* __restrict__ plm) {
  int idx = blockIdx.x * 256 + threadIdx.x;
  if (idx >= MMAX * NLAT) return;
  int m = idx / NLAT;
  int k = idx % NLAT;
  double theta = PI_D * (double)k / 511.0;
  double ct = cos(theta), st = sin(theta);
  double pmm = sqrt(1.0 / (4.0 * PI_D));
  for (int i = 1; i <= m; ++i)
    pmm *= -sqrt((2.0 * i + 1.0) / (2.0 * i)) * st;
  for (int l = 0; l < m; ++l)
    plm[((size_t)l * MMAX + m) * NLAT + k] = 0.0f;
  plm[((size_t)m * MMAX + m) * NLAT + k] = (float)pmm;
  double pprev = pmm;
  double pcur  = 0.0;
  if (m + 1 < LMAX) {
    pcur = sqrt(2.0 * m + 3.0) * ct * pmm;
    plm[((size_t)(m + 1) * MMAX + m) * NLAT + k] = (float)pcur;
  }
  for (int l = m + 2; l < LMAX; ++l) {
    double a = sqrt((4.0 * l * l - 1.0) / ((double)l * l - (double)m * m));
    double b = sqrt((((double)l - 1.0) * ((double)l - 1.0) - (double)m * m) /
                    (4.0 * ((double)l - 1.0) * ((double)l - 1.0) - 1.0));
    double pn = a * (ct * pcur - b * pprev);
    pprev = pcur; pcur = pn;
    plm[((size_t)l * MMAX + m) * NLAT + k] = (float)pn;
  }
}

// ---------------------------------------------------------------------------
// Branchless insert of packed key into running 3-min list (u32, monotonic).
// ---------------------------------------------------------------------------
__device__ __forceinline__ void ins3(unsigned kk, unsigned& k0, unsigned& k1,
                                     unsigned& k2) {
  bool c0 = kk < k0, c1 = kk < k1, c2 = kk < k2;
  unsigned n2 = c2 ? kk : k2;  n2 = c1 ? k1 : n2;
  unsigned n1 = c1 ? kk : k1;  n1 = c0 ? k0 : n1;
  k0 = c0 ? kk : k0;
  k1 = n1; k2 = n2;
}

// ---------------------------------------------------------------------------
// Stage 2: WMMA-based brute-force 3-NN + distance-weighted interpolation.
// Each wave handles 32 grid points (two 16-column B tiles sharing one A
// load); loops 128 tiles of 16 cloud points with the next A tile
// software-prefetched behind the insert chains.  Distances are clamped to
// >= 0 (the dot form can go epsilon-negative on near-coincident points,
// which would wreck the unsigned-key ordering) and packed into u32 keys
// (21 distance bits | 11-bit point index) so ordering matches top_k.
// Per-lane 3-min over the lane's half of the cloud rows; shfl_xor(16)
// merge; exact d2 recomputed for the 3 winners from LDS point data.
// ---------------------------------------------------------------------------
__global__ __launch_bounds__(256) void knn_interp_kernel(
    const float4* __restrict__ pts4, const float* __restrict__ rarr,
    float* __restrict__ interp) {
  __shared__ float4 spts[NPTS];   // 32 KB
  __shared__ float  sr[NPTS];     //  8 KB
  int bb = blockIdx.x >> 11;      // 2048 blocks per (branch,batch)
  int tg = blockIdx.x & 2047;
  const float4* gp = pts4 + (size_t)bb * NPTS;
  const float*  gr = rarr + (size_t)bb * NPTS;

  // LDS fill: async global->LDS (ASYNCcnt) when available.
#if HAVE_ASYNC_LDS
  for (int i = threadIdx.x; i < NPTS; i += 256) {
    __builtin_amdgcn_global_load_async_to_lds_b128(
        (gas_v4i*)(gp + i), (las_v4i*)(&spts[i]), 0, 0);
  }
  {
    const float4* gr4 = (const float4*)gr;
    float4* sr4 = (float4*)sr;
    for (int i = threadIdx.x; i < NPTS / 4; i += 256) {
      __builtin_amdgcn_global_load_async_to_lds_b128(
          (gas_v4i*)(gr4 + i), (las_v4i*)(&sr4[i]), 0, 0);
    }
  }
#if __has_builtin(__builtin_amdgcn_s_wait_asynccnt)
  __builtin_amdgcn_s_wait_asynccnt(0);
#else
  asm volatile("s_wait_asynccnt 0" ::: "memory");
#endif
#else
  for (int i = threadIdx.x; i < NPTS; i += 256) {
    spts[i] = gp[i];
    sr[i]   = gr[i];
  }
#endif
  __syncthreads();

  int lane   = threadIdx.x & 31;
  int wave   = threadIdx.x >> 5;
  int lane15 = lane & 15;
  int gbase  = tg * 256 + wave * 32;
  bool lo    = (lane < 16);

  // two grid columns per lane: g0 (tile 0) and g1 = g0 + 16 (tile 1)
  int g0 = gbase + lane15;
  int g1 = g0 + 16;
  float t0 = (float)(g0 >> 10) * (PI_F / 512.0f);
  float p0 = (float)((g0 & 1023) - 512) * (PI_F / 512.0f);
  float gg0 = t0 * t0 + p0 * p0;
  float t1 = (float)(g1 >> 10) * (PI_F / 512.0f);
  float p1 = (float)((g1 & 1023) - 512) * (PI_F / 512.0f);
  float gg1 = t1 * t1 + p1 * p1;
  // B layout: VGPR0 = K0 (lanes 0-15) / K2 (lanes 16-31); VGPR1 = K1 / K3
  v2f bmat0, bmat1;
  bmat0.x = lo ? t0 : gg0;  bmat0.y = lo ? p0 : 1.0f;
  bmat1.x = lo ? t1 : gg1;  bmat1.y = lo ? p1 : 1.0f;

  unsigned a0 = 0xFFFFFFFFu, a1 = 0xFFFFFFFFu, a2 = 0xFFFFFFFFu;  // tile 0
  unsigned b0 = 0xFFFFFFFFu, b1 = 0xFFFFFFFFu, b2 = 0xFFFFFFFFu;  // tile 1
  const float2* sp2 = reinterpret_cast<const float2*>(spts);
  int h = lane >> 4;

  // prefetch first A tile
  float2 acur = sp2[(size_t)lane15 * 2 + h];
  for (int nt = 0; nt < NPTS / 16; ++nt) {
    v2f amat; amat.x = acur.x; amat.y = acur.y;
    int ntn = (nt + 1) & (NPTS / 16 - 1);           // wraparound prefetch
    acur = sp2[(size_t)(ntn * 16 + lane15) * 2 + h];
    v8f c = {};
    v8f d0 = __builtin_amdgcn_wmma_f32_16x16x4_f32(
        false, amat, false, bmat0, (short)0, c, false, false);
    v8f d1 = __builtin_amdgcn_wmma_f32_16x16x4_f32(
        false, amat, false, bmat1, (short)0, c, false, false);
    int nb = nt * 16 + h * 8;   // cloud row base for this half-wave
#pragma unroll
    for (int v = 0; v < 8; ++v) {
      unsigned kk0 = (__float_as_uint(fmaxf(d0[v], 0.0f)) & 0xFFFFF800u) |
                     (unsigned)(nb + v);
      ins3(kk0, a0, a1, a2);
      unsigned kk1 = (__float_as_uint(fmaxf(d1[v], 0.0f)) & 0xFFFFF800u) |
                     (unsigned)(nb + v);
      ins3(kk1, b0, b1, b2);
    }
  }

  // merge the two half-wave 3-min lists (rows 0-7 vs 8-15)
  unsigned x;
  x = (unsigned)__shfl_xor((int)a0, 16, 32); ins3(x, a0, a1, a2);
  x = (unsigned)__shfl_xor((int)a1, 16, 32); ins3(x, a0, a1, a2);
  x = (unsigned)__shfl_xor((int)a2, 16, 32); ins3(x, a0, a1, a2);
  x = (unsigned)__shfl_xor((int)b0, 16, 32); ins3(x, b0, b1, b2);
  x = (unsigned)__shfl_xor((int)b1, 16, 32); ins3(x, b0, b1, b2);
  x = (unsigned)__shfl_xor((int)b2, 16, 32); ins3(x, b0, b1, b2);

  if (lo) {
    {
      unsigned ks[3] = {a0, a1, a2};
      float num = 0.0f, den = 0.0f;
#pragma unroll
      for (int t = 0; t < 3; ++t) {
        int ip = (int)(ks[t] & 2047u);
        float4 a = spts[ip];
        float d2 = fmaxf(fmaf(a.x, t0, fmaf(a.y, p0, gg0 + a.w)), 0.0f);
        num = fmaf(d2, sr[ip], num);
        den += d2;
      }
      interp[(size_t)bb * MGRID + g0] = num / den;
    }
    {
      unsigned ks[3] = {b0, b1, b2};
      float num = 0.0f, den = 0.0f;
#pragma unroll
      for (int t = 0; t < 3; ++t) {
        int ip = (int)(ks[t] & 2047u);
        float4 a = spts[ip];
        float d2 = fmaxf(fmaf(a.x, t1, fmaf(a.y, p1, gg1 + a.w)), 0.0f);
        num = fmaf(d2, sr[ip], num);
        den += d2;
      }
      interp[(size_t)bb * MGRID + g1] = num / den;
    }
  }
}

// ---------------------------------------------------------------------------
// Stage 3: cosine DFT (real part of rfft, first 50 modes) x CC weights.
// Fw[bb][m][k] = (2*pi/1024) * wq[k] * sum_n interp[bb][k][n] * cos(pi*m*n/512)
// ---------------------------------------------------------------------------
__global__ void dft_kernel(const float* __restrict__ interp,
                           const float* __restrict__ wq,
                           float* __restrict__ Fw) {
  int idx = blockIdx.x * 256 + threadIdx.x;   // bb*512*50 + k*50 + m
  if (idx >= NBB * NLAT * MMAX) return;
  int bb = idx / (NLAT * MMAX);
  int k  = (idx / MMAX) % NLAT;
  int m  = idx % MMAX;
  const float* row = interp + ((size_t)bb * NLAT + k) * NLON;
  float acc = 0.0f;
  for (int n = 0; n < NLON; ++n) {
    // argument of cos is pi * (m*n/512): exact in fp32 (m*n < 2^24, /512 exact)
    acc += row[n] * cospif((float)(m * n) * (1.0f / 512.0f));
  }
  Fw[((size_t)bb * MMAX + m) * NLAT + k] =
      acc * wq[k] * (2.0f * PI_F / (float)NLON);
}

// ---------------------------------------------------------------------------
// Stage 4: Legendre contraction  coeff[bb][l][m] = sum_k Fw[bb][m][k]*P[l][m][k]
// ---------------------------------------------------------------------------
__global__ void einsum_kernel(const float* __restrict__ Fw,
                              const float* __restrict__ plm,
                              float* __restrict__ coeff) {
  int idx = blockIdx.x * 256 + threadIdx.x;   // bb*2500 + l*50 + m
  if (idx >= NBB * LMAX * MMAX) return;
  int bb = idx / (LMAX * MMAX);
  int l  = (idx / MMAX) % LMAX;
  int m  = idx % MMAX;
  const float* fr = Fw + ((size_t)bb * MMAX + m) * NLAT;
  const float* pr = plm + ((size_t)l * MMAX + m) * NLAT;
  float acc = 0.0f;
  for (int k = 0; k < NLAT; ++k) acc += fr[k] * pr[k];
  coeff[idx] = acc;
}

// ---------------------------------------------------------------------------
// Stage 5: loss = mean over (b,l,m) of (coeff_pred - coeff_target)^2
// ---------------------------------------------------------------------------
__global__ void loss_kernel(const float* __restrict__ coeff,
                            float* __restrict__ out) {
  __shared__ float sbuf[256];
  float acc = 0.0f;
  for (int i = threadIdx.x; i < 2 * LMAX * MMAX; i += 256) {
    int b  = i / (LMAX * MMAX);
    int lm = i % (LMAX * MMAX);
    float cp = coeff[(size_t)(0 + b) * (LMAX * MMAX) + lm];
    float ct = coeff[(size_t)(2 + b) * (LMAX * MMAX) + lm];
    float dd = cp - ct;
    acc += dd * dd;
  }
  sbuf[threadIdx.x] = acc;
  __syncthreads();
  for (int s = 128; s > 0; s >>= 1) {
    if ((int)threadIdx.x < s) sbuf[threadIdx.x] += sbuf[threadIdx.x + s];
    __syncthreads();
  }
  if (threadIdx.x == 0) out[0] = sbuf[0] / (float)(2 * LMAX * MMAX);
}

// ---------------------------------------------------------------------------
extern "C" void kernel_launch(void* const* d_in, const int* in_sizes, int n_in,
                              void* d_out, int out_size, void* d_ws,
                              size_t ws_size, hipStream_t stream) {
  const float* pred   = (const float*)d_in[0];
  const float* target = (const float*)d_in[1];
  float* ws = (float*)d_ws;

  // workspace layout (floats)
  float4* pts4  = (float4*)ws;          //      0 .. 32768   (4*2048*4)
  float* rr     = ws + 32768;           //  32768 .. 40960
  float* wq     = ws + 40960;           //  40960 .. 41472
  float* plm    = ws + 41472;           //  41472 .. 1321472  (50*50*512)
  float* interp = ws + 1321472;         // .. 3418624         (4*524288)
  float* Fw     = ws + 3418624;         // .. 3521024         (4*50*512)
  float* coeff  = ws + 3521024;         // .. 3531024         (4*2500)

  prep_kernel<<<(NBB * NPTS + 255) / 256, 256, 0, stream>>>(pred, target, pts4, rr);
  wq_kernel<<<(NLAT + 255) / 256, 256, 0, stream>>>(wq);
  plm_kernel<<<(MMAX * NLAT + 255) / 256, 256, 0, stream>>>(plm);
  knn_interp_kernel<<<NBB * (MGRID / 256), 256, 0, stream>>>(pts4, rr, interp);
  dft_kernel<<<(NBB * NLAT * MMAX + 255) / 256, 256, 0, stream>>>(interp, wq, Fw);
  einsum_kernel<<<(NBB * LMAX * MMAX + 255) / 256, 256, 0, stream>>>(Fw, plm, coeff);
  loss_kernel<<<1, 256, 0, stream>>>(coeff, (float*)d_out);
}